// AdditiveAttention_18554258718776
// MI455X (gfx1250) — compile-verified
//
#include <hip/hip_runtime.h>

// Additive (Bahdanau) attention for MI455X / gfx1250, wave32.
// B=8, LQ=128, LK=1024, QD=KD=VD=H=128.
//
// Phase 1: q = Q@W_q, k = K@W_k          -> fp32 WMMA 16x16x4 GEMM
// Phase 2 (fused, per (b, 16-row q-tile)):
//   scores = wv . tanh(q[l]+k[s])        -> v_tanh_f32 (trans), masked-skip
//   softmax over LK (half-wave shuffles)
//   out = attn @ V                       -> fp32 WMMA 16x16x4

typedef __attribute__((ext_vector_type(2))) float v2f;
typedef __attribute__((ext_vector_type(8))) float v8f;

#if defined(__has_builtin)
#  if __has_builtin(__builtin_amdgcn_tanhf)
#    define TANHF(x) __builtin_amdgcn_tanhf(x)   // v_tanh_f32 on gfx1250
#  else
#    define TANHF(x) tanhf(x)
#  endif
#else
#  define TANHF(x) tanhf(x)
#endif

#define AA_B   8
#define AA_LQ  128
#define AA_LK  1024
#define AA_H   128
#define AA_VD  128
#define AA_NEG_INF (-1e6f)

// ---------------------------------------------------------------------------
// fp32 WMMA wrapper: D = A(16x4) * B(4x16) + C, per ISA 7.12.2 layouts.
//   A frag (v2f): lane L -> row M = L%16, K = {kb, kb+1}, kb = kk + 2*(L/16)
//   B frag (v2f): lane L -> col N = L%16, K = {kb, kb+1}
//   C/D  (v8f):  vgpr r -> M = r + 8*(L/16), N = L%16
// ---------------------------------------------------------------------------
__device__ __forceinline__ v8f wmma_f32_4(v2f a, v2f b, v8f c) {
  return __builtin_amdgcn_wmma_f32_16x16x4_f32(
      /*neg_a=*/false, a, /*neg_b=*/false, b,
      /*c_mod=*/(short)0, c, /*reuse_a=*/false, /*reuse_b=*/false);
}

// ---------------------------------------------------------------------------
// Kernel 1: Y[nrows,128] = X[nrows,128] @ W[128,128], fp32 WMMA.
// Block = 256 threads (8 waves) -> 16 rows x 128 cols; wave w owns cols 16w..
// ---------------------------------------------------------------------------
__global__ __launch_bounds__(256)
void proj_gemm_wmma(const float* __restrict__ X, const float* __restrict__ W,
                    float* __restrict__ Y) {
  __shared__ float As[16][132];                 // pad 132: conflict-free frags
  const int row0 = blockIdx.x * 16;

  for (int i = threadIdx.x; i < 16 * AA_H; i += 256) {
    const int r = i >> 7, c = i & 127;
    As[r][c] = X[(size_t)(row0 + r) * AA_H + c];
  }
  __syncthreads();

  const int lane  = threadIdx.x & 31;
  const int n0    = (threadIdx.x >> 5) << 4;    // wave -> 16-col tile
  const int mrow  = lane & 15;
  const int khalf = lane >> 4;

  v8f acc = {};
  #pragma unroll 4
  for (int kk = 0; kk < AA_H; kk += 4) {
    const int kb = kk + 2 * khalf;
    v2f a, b;
    a.x = As[mrow][kb];
    a.y = As[mrow][kb + 1];
    b.x = W[(size_t)kb * AA_H + n0 + mrow];
    b.y = W[(size_t)(kb + 1) * AA_H + n0 + mrow];
    acc = wmma_f32_4(a, b, acc);
  }

  #pragma unroll
  for (int r = 0; r < 8; ++r)
    Y[(size_t)(row0 + r + 8 * khalf) * AA_H + n0 + mrow] = acc[r];
}

// ---------------------------------------------------------------------------
// Kernel 2: fused scores -> mask -> softmax -> attn@V.
// Grid: (LQ/16, B). Block: 256 threads (8 waves).
//   thread t: l = t>>4 (query row in tile), si = t&15, s = si + 16*j.
//   Threads sharing l are one half-wave -> q_s reads broadcast, shuffle reduce.
// ---------------------------------------------------------------------------
__global__ __launch_bounds__(256)
void attn_fused(const float* __restrict__ qh, const float* __restrict__ kh,
                const float* __restrict__ V, const int* __restrict__ valid_lens,
                const float* __restrict__ wv, float* __restrict__ out) {
  __shared__ float q_s[16][132];                // 8.4 KB
  __shared__ float p_s[16][AA_LK + 1];          // 65.6 KB, stride 1025 -> no conflicts

  const int b    = blockIdx.y;
  const int row0 = blockIdx.x * 16;
  const float* qb = qh + ((size_t)b * AA_LQ + row0) * AA_H;
  const float* kb = kh + (size_t)b * AA_LK * AA_H;
  const float* Vb = V  + (size_t)b * AA_LK * AA_VD;
  const int valid = valid_lens[b];

  for (int i = threadIdx.x; i < 16 * AA_H; i += 256) {
    const int r = i >> 7, c = i & 127;
    q_s[r][c] = qb[(size_t)r * AA_H + c];
  }
  __syncthreads();

  const int l  = threadIdx.x >> 4;              // 0..15
  const int si = threadIdx.x & 15;              // 0..15

  // ---- scores (dominant tanh work; skip masked keys entirely) ----
  for (int j = 0; j < AA_LK / 16; ++j) {
    const int s = si + 16 * j;
    float score = AA_NEG_INF;
    if (s < valid) {
      const float* krow = kb + (size_t)s * AA_H;
      float acc = 0.f;
      #pragma unroll
      for (int h = 0; h < AA_H; h += 4) {
        const float4 kv = *reinterpret_cast<const float4*>(krow + h);
        const float4 qv = *reinterpret_cast<const float4*>(&q_s[l][h]);
        const float4 w  = *reinterpret_cast<const float4*>(wv + h);  // uniform -> s_load
        acc += w.x * TANHF(qv.x + kv.x);
        acc += w.y * TANHF(qv.y + kv.y);
        acc += w.z * TANHF(qv.z + kv.z);
        acc += w.w * TANHF(qv.w + kv.w);
      }
      score = acc;
    }
    p_s[l][s] = score;
  }
  __syncthreads();

  // ---- softmax per row: 16 lanes (one half-wave) cooperate ----
  float mx = AA_NEG_INF;
  for (int s = si; s < AA_LK; s += 16) mx = fmaxf(mx, p_s[l][s]);
  #pragma unroll
  for (int off = 8; off > 0; off >>= 1) mx = fmaxf(mx, __shfl_xor(mx, off, 16));

  float sum = 0.f;
  for (int s = si; s < AA_LK; s += 16) {
    const float e = __expf(p_s[l][s] - mx);     // masked -> exp(-1e6-mx) == 0
    p_s[l][s] = e;
    sum += e;
  }
  #pragma unroll
  for (int off = 8; off > 0; off >>= 1) sum += __shfl_xor(sum, off, 16);

  const float inv = 1.0f / sum;
  for (int s = si; s < AA_LK; s += 16) p_s[l][s] *= inv;
  __syncthreads();

  // ---- out = attn @ V via fp32 WMMA; wave w owns 16-col tile ----
  const int lane  = threadIdx.x & 31;
  const int n0    = (threadIdx.x >> 5) << 4;
  const int mrow  = lane & 15;
  const int khalf = lane >> 4;

  v8f acc = {};
  #pragma unroll 2
  for (int kk = 0; kk < AA_LK; kk += 4) {
    const int kv = kk + 2 * khalf;
    v2f a, bf;
    a.x  = p_s[mrow][kv];
    a.y  = p_s[mrow][kv + 1];
    bf.x = Vb[(size_t)kv * AA_VD + n0 + mrow];
    bf.y = Vb[(size_t)(kv + 1) * AA_VD + n0 + mrow];
    acc = wmma_f32_4(a, bf, acc);
  }

  float* ob = out + ((size_t)b * AA_LQ + row0) * AA_VD;
  #pragma unroll
  for (int r = 0; r < 8; ++r)
    ob[(size_t)(r + 8 * khalf) * AA_VD + n0 + mrow] = acc[r];
}

// ---------------------------------------------------------------------------
extern "C" void kernel_launch(void* const* d_in, const int* in_sizes, int n_in,
                              void* d_out, int out_size, void* d_ws, size_t ws_size,
                              hipStream_t stream) {
  (void)in_sizes; (void)n_in; (void)out_size; (void)ws_size;
  const float* queries = (const float*)d_in[0];   // [B,LQ,QD]
  const float* keys    = (const float*)d_in[1];   // [B,LK,KD]
  const float* values  = (const float*)d_in[2];   // [B,LK,VD]
  const int*   vlens   = (const int*)  d_in[3];   // [B]
  const float* Wq      = (const float*)d_in[4];   // [QD,H]
  const float* Wk      = (const float*)d_in[5];   // [KD,H]
  const float* wv      = (const float*)d_in[6];   // [H]
  float* out = (float*)d_out;                     // [B,LQ,VD]

  float* qh = (float*)d_ws;                       // [B*LQ, H]   = 512 KB
  float* kh = qh + (size_t)AA_B * AA_LQ * AA_H;   // [B*LK, H]   = 4 MB

  proj_gemm_wmma<<<(AA_B * AA_LQ) / 16, 256, 0, stream>>>(queries, Wq, qh);
  proj_gemm_wmma<<<(AA_B * AA_LK) / 16, 256, 0, stream>>>(keys,    Wk, kh);
  attn_fused<<<dim3(AA_LQ / 16, AA_B), 256, 0, stream>>>(qh, kh, values, vlens, wv, out);
}